// Attention_27848567947458
// MI455X (gfx1250) — compile-verified
//
#include <hip/hip_runtime.h>

// CDNA5 / gfx1250 fused attention:
//   scores = (Q K^T) * 1/sqrt(D)  -> row softmax -> p_attn (written) -> O = P V
// fp32 end-to-end using V_WMMA_F32_16X16X4_F32 (wave32 matrix core, fp32 in/out).
// K/V tiles staged to LDS with GLOBAL_LOAD_ASYNC_TO_LDS_B128 (ASYNCcnt path),
// p_attn streamed with non-temporal b128 stores.

typedef float v2f __attribute__((ext_vector_type(2)));
typedef float v4f __attribute__((ext_vector_type(4)));
typedef float v8f __attribute__((ext_vector_type(8)));

#define S_LEN 2048
#define D_DIM 64
#define H_NUM 16
#define B_NUM 2

#define MTILE 16          // query rows per workgroup (one WMMA M tile)
#define KTILE 64          // keys/values staged per LDS tile
#define BLOCK 128         // 4 waves of 32

#define PSTR 2052         // score-row stride (floats): +4 banks/row, 16B-aligned rows
#define TSTR 72           // K/V tile row stride: +8 banks/row, 16B-aligned rows
#define QSTR 68           // Q tile row stride

// Async global->LDS copy of one 64x64 f32 tile (16 KB): each of the 128 threads
// issues 8 b128 transfers that land directly in LDS (no VGPR round trip).
// GVS addressing: mem = SGPR64_base + VGPR32_offset; VDST = per-lane LDS byte addr.
__device__ __forceinline__ void async_tile_load(const float* gsrc, float* ldst, int tid)
{
    const unsigned long long gbase = (unsigned long long)(uintptr_t)gsrc;
    #pragma unroll
    for (int ii = 0; ii < 8; ++ii) {
        const int i = tid + ii * BLOCK;
        const int r = i >> 4, c = (i & 15) << 2;
        const unsigned lds  = (unsigned)(uintptr_t)&ldst[r * TSTR + c];
        const unsigned goff = (unsigned)((r * D_DIM + c) * sizeof(float));
        asm volatile("global_load_async_to_lds_b128 %0, %1, %2"
                     :: "v"(lds), "v"(goff), "s"(gbase) : "memory");
    }
    asm volatile("s_wait_asynccnt 0x0" ::: "memory");
}

__global__ __launch_bounds__(BLOCK)
void attn_fused_f32wmma(const float* __restrict__ Q,
                        const float* __restrict__ K,
                        const float* __restrict__ V,
                        float* __restrict__ Out,
                        float* __restrict__ Pout)
{
    extern __shared__ float smem[];
    float* sP   = smem;                    // MTILE * PSTR   (exp scores, unnormalized)
    float* sT   = sP + MTILE * PSTR;       // KTILE * TSTR   (K tile, then V tiles)
    float* sQ   = sT + KTILE * TSTR;       // MTILE * QSTR
    float* sInv = sQ + MTILE * QSTR;       // MTILE          (1/rowsum)

    const int tid  = threadIdx.x;
    const int lane = tid & 31;
    const int w    = tid >> 5;             // wave id 0..3
    const int lm   = lane & 15;            // lane % 16
    const int lh   = lane >> 4;            // lane half (0/1)

    const int q0 = blockIdx.x * MTILE;
    const size_t bh = (size_t)blockIdx.z * H_NUM + blockIdx.y;

    const float* Qb = Q + (bh * S_LEN + q0) * D_DIM;
    const float* Kb = K + bh * S_LEN * D_DIM;
    const float* Vb = V + bh * S_LEN * D_DIM;

    // ---- stage Q tile (16 x 64) ----
    for (int i = tid; i < MTILE * (D_DIM / 4); i += BLOCK) {
        int r = i >> 4, c = (i & 15) << 2;
        *(v4f*)&sQ[r * QSTR + c] = *(const v4f*)&Qb[r * D_DIM + c];
    }
    __syncthreads();

    // A-operand (Q) held in registers for the whole kernel:
    // f32 WMMA A layout: lane holds A[lane%16][2*(lane/16) + {0,1}] per K-step of 4.
    v2f aq[16];
    #pragma unroll
    for (int s = 0; s < 16; ++s)
        aq[s] = *(const v2f*)&sQ[lm * QSTR + 4 * s + 2 * lh];

    const float scale = 0.125f;            // 1/sqrt(64)

    // ================= phase 1: scores = scale * Q K^T =================
    for (int kt = 0; kt < S_LEN / KTILE; ++kt) {
        __syncthreads();                   // previous sT consumed by all waves
        async_tile_load(Kb + kt * KTILE * D_DIM, sT, tid);
        __syncthreads();

        v8f acc = {};
        #pragma unroll
        for (int s = 0; s < 16; ++s) {
            // B = K^T: lane needs K[key = w*16 + lane%16][d = 4s + 2*(lane/16) + {0,1}]
            v2f bv = *(const v2f*)&sT[(w * 16 + lm) * TSTR + 4 * s + 2 * lh];
            acc = __builtin_amdgcn_wmma_f32_16x16x4_f32(
                      false, aq[s], false, bv, (short)0, acc, false, false);
        }
        const int colbase = kt * KTILE + w * 16 + lm;
        #pragma unroll
        for (int j = 0; j < 8; ++j)        // C layout: row = j + 8*lh, col = lane%16
            sP[(j + 8 * lh) * PSTR + colbase] = acc[j] * scale;
    }
    __syncthreads();

    // ================= phase 2: row softmax (8 threads per row) =================
    {
        const int r  = tid >> 3;
        float* row   = &sP[r * PSTR + ((tid & 7) << 8)];
        float m = -__builtin_inff();
        #pragma unroll 4
        for (int i = 0; i < 64; ++i) {
            v4f v = *(v4f*)&row[i * 4];
            m = fmaxf(m, fmaxf(fmaxf(v.x, v.y), fmaxf(v.z, v.w)));
        }
        m = fmaxf(m, __shfl_xor(m, 1, 8));
        m = fmaxf(m, __shfl_xor(m, 2, 8));
        m = fmaxf(m, __shfl_xor(m, 4, 8));
        float sum = 0.f;
        #pragma unroll 4
        for (int i = 0; i < 64; ++i) {
            v4f v = *(v4f*)&row[i * 4];
            v.x = __expf(v.x - m); v.y = __expf(v.y - m);
            v.z = __expf(v.z - m); v.w = __expf(v.w - m);
            *(v4f*)&row[i * 4] = v;        // keep UNnormalized exp in LDS
            sum += (v.x + v.y) + (v.z + v.w);
        }
        sum += __shfl_xor(sum, 1, 8);
        sum += __shfl_xor(sum, 2, 8);
        sum += __shfl_xor(sum, 4, 8);
        if ((tid & 7) == 0) sInv[r] = 1.0f / sum;
    }
    __syncthreads();

    // ================= phase 3: stream p_attn (dominant traffic, NT stores) =========
    {
        float* Pb = Pout + (bh * S_LEN + q0) * (size_t)S_LEN;
        for (int r = 0; r < MTILE; ++r) {
            const float inv = sInv[r];
            float* dst = Pb + (size_t)r * S_LEN;
            for (int i = tid; i < S_LEN / 4; i += BLOCK) {
                v4f v = *(v4f*)&sP[r * PSTR + i * 4];
                v = v * inv;
                __builtin_nontemporal_store(v, (v4f*)&dst[i * 4]);
            }
        }
    }

    // ================= phase 4: O = (exp-scores) * V, rescaled by 1/sum ============
    v8f oacc = {};
    for (int kt = 0; kt < S_LEN / KTILE; ++kt) {
        __syncthreads();                   // sT reuse
        async_tile_load(Vb + kt * KTILE * D_DIM, sT, tid);
        __syncthreads();

        #pragma unroll
        for (int s = 0; s < 16; ++s) {
            // A = P: contiguous v2f straight out of the score buffer (no transpose needed)
            v2f av = *(const v2f*)&sP[lm * PSTR + kt * KTILE + 4 * s + 2 * lh];
            const int krow = 4 * s + 2 * lh;
            v2f bv;                        // B = V[k][d], wave w covers d-cols w*16..+15
            bv.x = sT[(krow    ) * TSTR + w * 16 + lm];
            bv.y = sT[(krow + 1) * TSTR + w * 16 + lm];
            oacc = __builtin_amdgcn_wmma_f32_16x16x4_f32(
                       false, av, false, bv, (short)0, oacc, false, false);
        }
    }

    float* Ob = Out + (bh * S_LEN + q0) * D_DIM;
    #pragma unroll
    for (int j = 0; j < 8; ++j) {
        const int m = j + 8 * lh;
        Ob[m * D_DIM + w * 16 + lm] = oacc[j] * sInv[m];
    }
}

extern "C" void kernel_launch(void* const* d_in, const int* in_sizes, int n_in,
                              void* d_out, int out_size, void* d_ws, size_t ws_size,
                              hipStream_t stream) {
    const float* Q = (const float*)d_in[0];
    const float* K = (const float*)d_in[1];
    const float* V = (const float*)d_in[2];
    float* Out  = (float*)d_out;                                   // [B,H,S,D]
    float* Pout = Out + (size_t)B_NUM * H_NUM * S_LEN * D_DIM;     // [B,H,S,S]

    dim3 grid(S_LEN / MTILE, H_NUM, B_NUM);                        // 128 x 16 x 2
    size_t shmem = (size_t)(MTILE * PSTR + KTILE * TSTR + MTILE * QSTR + MTILE)
                   * sizeof(float);                                // ~150.6 KB -> 2 WG/WGP
    hipLaunchKernelGGL(attn_fused_f32wmma, grid, dim3(BLOCK), shmem, stream,
                       Q, K, V, Out, Pout);
}